// Diffusion1D_8881992368812
// MI455X (gfx1250) — compile-verified
//
#include <hip/hip_runtime.h>
#include <cstdint>

// ---- tiling -----------------------------------------------------------------
#define WIN   4096                 // window elements per workgroup (16 KB LDS)
#define HALO  512                  // halo each side: mu^512 ~ 6e-23 for dt=0.1
#define SOUT  (WIN - 2 * HALO)     // 3072 outputs per workgroup
#define TPT   16                   // elements per thread
#define BLOCK 256                  // 8 wave32 waves

typedef unsigned int u32x4 __attribute__((ext_vector_type(4)));
typedef int          i32x8 __attribute__((ext_vector_type(8)));
typedef int          i32x4 __attribute__((ext_vector_type(4)));

__global__ __launch_bounds__(BLOCK)
void diffusion1d_window_kernel(const float* __restrict__ C,
                               const float* __restrict__ dtp,
                               const float* __restrict__ csp,
                               const float* __restrict__ cbp,
                               float* __restrict__ out, int n)
{
  __shared__ __align__(16) float sf[WIN];
  __shared__ float waveP[8];
  __shared__ float waveQ[8];
  __shared__ float sEdge;

  const int tid  = threadIdx.x;
  const int lane = tid & 31;
  const int wv   = tid >> 5;
  const int blk  = blockIdx.x;
  const int nblk = gridDim.x;
  const int wb   = blk * SOUT - HALO;      // window base (global i), fits int32

  // ---- uniform scalars / Toeplitz closed form -------------------------------
  const float dtv = dtp[0];
  const float Cs  = csp[0];
  const float Cb  = cbp[0];
  const double r   = (1e-05 / (1e-04 * 1e-04)) * (double)dtv; // D*dt/dx^2
  const double kap = sqrt(1.0 + 4.0 * r);
  const double mud = (2.0 * r) / ((1.0 + 2.0 * r) + kap);     // decay root, 0<=mu<1
  const float  mu   = (float)mud;
  const float  rf   = (float)r;
  const float  invk = (float)(1.0 / kap);
  const float  lnmu = (float)log(mud);                        // -inf if r==0 (ok)
  double p1, p2, p4, p8, p16;
  { double m2 = mud * mud, m4 = m2 * m2, m8 = m4 * m4;
    p1 = m8 * m8; p2 = p1 * p1; p4 = p2 * p2; p8 = p4 * p4; p16 = p8 * p8; }
  const float phi[5] = { (float)p1, (float)p2, (float)p4, (float)p8, (float)p16 };

  // ---- window extent (clamped to [0,n)) -------------------------------------
  const int startE  = (wb < 0) ? 0 : wb;
  const int ldsOffE = startE - wb;
  const int remain  = n - startE;
  int tileE = WIN - ldsOffE;
  if (tileE > remain) tileE = remain;

  // ---- Phase A: async tensor DMA (TDM) global -> LDS ------------------------
#if defined(__HIP_DEVICE_COMPILE__) && __has_builtin(__builtin_amdgcn_tensor_load_to_lds)
  if (wv == 0) {
    const unsigned ldsA = (unsigned)(uintptr_t)(void*)sf + (unsigned)(ldsOffE * 4);
    const unsigned long long ga = (unsigned long long)(uintptr_t)(C + startE);
    const unsigned td0 = (unsigned)tileE;
    u32x4 g0;                               // D# group 0 (ISA 08 sec 8.3)
    g0[0] = 1u;                             // count=1 valid descriptor
    g0[1] = ldsA;                           // lds_addr
    g0[2] = (unsigned)(ga & 0xFFFFFFFFu);   // global_addr[31:0]
    g0[3] = (unsigned)((ga >> 32) & 0x01FFFFFFu) | (2u << 30); // addr[56:32], type=2
    i32x8 g1;                               // D# group 1 (sec 8.4)
    g1[0] = (int)(2u << 16);                // wg_mask=0, data_size=2 (4 bytes)
    g1[1] = (int)((td0 & 0xFFFFu) << 16);   // tensor_dim0[15:0]
    g1[2] = (int)(((td0 >> 16) & 0xFFFFu) | (1u << 16)); // dim0[31:16], tensor_dim1=1
    g1[3] = (int)((td0 & 0xFFFFu) << 16);   // tile_dim0 (<=4096 fits 16b)
    g1[4] = 1;                              // tile_dim1=1
    g1[5] = (int)td0;                       // tensor_dim0_stride[31:0]
    g1[6] = 0;
    g1[7] = 0;
    i32x4 z4 = {0, 0, 0, 0};
  #if __has_include(<hip/amd_detail/amd_gfx1250_TDM.h>)
    i32x8 z8 = {0, 0, 0, 0, 0, 0, 0, 0};    // clang-23 / therock headers: 6-arg form
    __builtin_amdgcn_tensor_load_to_lds(g0, g1, z4, z4, z8, 0);
  #else
    __builtin_amdgcn_tensor_load_to_lds(g0, g1, z4, z4, 0); // ROCm 7.2 5-arg form
  #endif
    __builtin_amdgcn_s_wait_tensorcnt(0);
  }
#else
  for (int idx = tid; idx < tileE; idx += BLOCK)
    sf[ldsOffE + idx] = C[startE + idx];
#endif
  __syncthreads();

  // ---- Phase B: per-thread RHS f_i (zero outside [1,n-2], + BC terms) -------
  const int kBase = tid * TPT;
  const int giBase = wb + kBase;
  float raw[TPT];
  float4* sf4 = (float4*)sf;
  #pragma unroll
  for (int q = 0; q < 4; ++q) {
    float4 v = sf4[tid * 4 + q];
    raw[q * 4 + 0] = v.x; raw[q * 4 + 1] = v.y;
    raw[q * 4 + 2] = v.z; raw[q * 4 + 3] = v.w;
  }
  #pragma unroll
  for (int j = 0; j < TPT; ++j) {
    const int gi = giBase + j;
    float f = (gi >= 1 && gi <= n - 2) ? raw[j] : 0.0f;
    if (gi == 1)     f = fmaf(rf, Cs, f);  // fold x0 = C_surf into RHS
    if (gi == n - 2) f = fmaf(rf, Cb, f);  // fold x[n-1] = C_bulk into RHS
    raw[j] = f;
  }

  // ---- Phase C: local geometric scans (fwd & bwd), 16 serial FMAs each ------
  const int kR = (n - 2) - wb;             // window pos of i = n-2 (last block)
  float x[TPT];
  float s = 0.0f, stashF = 0.0f, stashB = 0.0f;
  int jR = 0;
  #pragma unroll
  for (int j = 0; j < TPT; ++j) {
    s = fmaf(mu, s, raw[j]);
    x[j] = s;
    if (kBase + j == kR) { stashF = s; jR = j; }
  }
  float Pt = s;
  s = 0.0f;
  #pragma unroll
  for (int j = TPT - 1; j >= 0; --j) {
    s = fmaf(mu, s, raw[j]);
    x[j] += s - raw[j];                    // x = fwd_local + bwd_local - f
    if (kBase + j == HALO + 1) stashB = s;
  }
  float Qt = s;

  // ---- Phase D: wave32 Hillis-Steele scans of thread totals (coeff mu^16) ---
  float P = Pt, Q = Qt;
  #pragma unroll
  for (int o = 0; o < 5; ++o) {
    const int off = 1 << o;
    const float pu = __shfl_up(P, off, 32);
    const float qd = __shfl_down(Q, off, 32);
    if (lane >= off)     P = fmaf(phi[o], pu, P);
    if (lane < 32 - off) Q = fmaf(phi[o], qd, Q);
  }
  if (lane == 31) waveP[wv] = P;
  if (lane == 0)  waveQ[wv] = Q;
  __syncthreads();
  // cross-wave carries (coupling beyond the adjacent wave is mu^512 ~ 0)
  const float carryW  = (wv > 0) ? waveP[wv - 1] : 0.0f;
  const float carryWB = (wv < 7) ? waveQ[wv + 1] : 0.0f;
  P = fmaf(__expf(lnmu * (float)(16 * (lane + 1))), carryW,  P);
  Q = fmaf(__expf(lnmu * (float)(16 * (32 - lane))), carryWB, Q);
  const float pu1 = __shfl_up(P, 1, 32);
  const float qd1 = __shfl_down(Q, 1, 32);
  const float cin  = (lane == 0)  ? carryW  : pu1;  // full Fwd carry into thread
  const float cinB = (lane == 31) ? carryWB : qd1;  // full Bwd carry into thread

  float pw = mu;
  #pragma unroll
  for (int j = 0; j < TPT; ++j) { x[j] = fmaf(pw, cin, x[j]);  pw *= mu; }
  pw = mu;
  #pragma unroll
  for (int j = TPT - 1; j >= 0; --j) { x[j] = fmaf(pw, cinB, x[j]); pw *= mu; }

  // ---- Phase E: analytic Dirichlet image corrections (boundary blocks only) -
  // x_i = invk * (Fwd_i + Bwd_i - f_i - mu^{i+1}*Bwd_1 - mu^{n-i}*Fwd_{n-2})
  // NOTE: correction subtracted UNSCALED here; invk is applied once below.
  const bool firstB = (blk == 0);
  const bool lastB  = (blk == nblk - 1);
  if (firstB || lastB) {
    if (firstB && kBase <= HALO + 1 && HALO + 1 < kBase + TPT)
      sEdge = fmaf(__expf(lnmu * 15.0f), cinB, stashB);          // Bwd at i=1
    if (lastB && kR >= kBase && kR < kBase + TPT)
      sEdge = fmaf(__expf(lnmu * (float)(jR + 1)), cin, stashF); // Fwd at i=n-2
    __syncthreads();
    const float e = sEdge;
    #pragma unroll
    for (int j = 0; j < TPT; ++j) {
      const int gi = giBase + j;
      const float dist = firstB ? (float)(gi + 1) : (float)(n - gi);
      x[j] -= e * __expf(lnmu * dist);     // fixed: no extra invk factor
    }
  }
  #pragma unroll
  for (int j = 0; j < TPT; ++j) x[j] *= invk;

  // ---- Phase F: LDS bounce -> fully coalesced b128 stores -------------------
  #pragma unroll
  for (int q = 0; q < 4; ++q)
    sf4[tid * 4 + q] = make_float4(x[q * 4 + 0], x[q * 4 + 1],
                                   x[q * 4 + 2], x[q * 4 + 3]);
  __syncthreads();
  #pragma unroll
  for (int m = 0; m < 3; ++m) {
    const int k2 = HALO + m * 1024 + tid * 4;      // outputs [HALO, HALO+SOUT)
    const int gi = wb + k2;                        // multiple of 4, >= 0
    if (gi < n) {                                  // n % 4 == 0 -> all-or-nothing
      float4 v = sf4[k2 >> 2];
      if (gi == 0)         v.x = Cs;               // Dirichlet x0
      if (gi + 3 == n - 1) v.w = Cb;               // Dirichlet x[n-1]
      ((float4*)out)[gi >> 2] = v;
    }
  }
}

extern "C" void kernel_launch(void* const* d_in, const int* in_sizes, int n_in,
                              void* d_out, int out_size, void* d_ws, size_t ws_size,
                              hipStream_t stream) {
  const float* C  = (const float*)d_in[0];
  const float* dt = (const float*)d_in[1];
  const float* cs = (const float*)d_in[2];
  const float* cb = (const float*)d_in[3];
  float* out = (float*)d_out;
  const int n = in_sizes[0];
  const int nblk = (n + SOUT - 1) / SOUT;
  diffusion1d_window_kernel<<<nblk, BLOCK, 0, stream>>>(C, dt, cs, cb, out, n);
  (void)d_ws; (void)ws_size; (void)n_in; (void)out_size;
}